// Readout_90993177133441
// MI455X (gfx1250) — compile-verified
//
#include <hip/hip_runtime.h>

#define HIDDEN 256
#define NCLS   128
#define BROWS  128   // node rows per block
#define KC     32    // K chunk staged in LDS
#define LSTR   34    // padded LDS stride (floats) per weight column

typedef float v2f __attribute__((ext_vector_type(2)));
typedef float v8f __attribute__((ext_vector_type(8)));

// ---------------------------------------------------------------------------
// Fused: gate_logits = Ain@Win + Aout@Wout + gate_b ; gating = sigmoid(...)
//        nodewise    = gating * (Aout@Wt + trans_b)
// One wave computes a 16(M) x 128(N) strip via 8+8 WMMA C-tiles (f32 16x16x4).
// Weights are staged per-K-chunk in LDS, transposed with padded stride so a
// B fragment is one conflict-free ds_load_b64 in native WMMA B layout.
// ---------------------------------------------------------------------------
__launch_bounds__(256)
__global__ void readout_fused(const float* __restrict__ Ain,
                              const float* __restrict__ Aout,
                              const float* __restrict__ gate_w,
                              const float* __restrict__ gate_b,
                              const float* __restrict__ trans_w,
                              const float* __restrict__ trans_b,
                              float* __restrict__ out_node,
                              int n_nodes)
{
    __shared__ float s_win [NCLS * LSTR];
    __shared__ float s_wout[NCLS * LSTR];
    __shared__ float s_wt  [NCLS * LSTR];

    const int tid   = threadIdx.x;
    const int lane  = tid & 31;
    const int wave  = tid >> 5;              // 0..7
    const int mrow  = lane & 15;             // M row (A) / N col (B,C) in tile
    const int khalf = (lane >> 4) << 1;      // 0 or 2: K pair selected by lane half
    const int rhalf = (lane >> 4) << 3;      // 0 or 8: C-row offset of lane half

    const int waveRow = blockIdx.x * BROWS + wave * 16;
    int arow = waveRow + mrow;
    if (arow >= n_nodes) arow = n_nodes - 1; // clamp; masked at store time

    const float* ainp  = Ain  + (size_t)arow * HIDDEN;
    const float* aoutp = Aout + (size_t)arow * HIDDEN;

    v8f accG[8], accT[8];
#pragma unroll
    for (int t = 0; t < 8; ++t) { accG[t] = (v8f)0.0f; accT[t] = (v8f)0.0f; }

    for (int kc = 0; kc < HIDDEN; kc += KC) {
        __syncthreads();
        // Stage weight K-chunks transposed: s[col * LSTR + k_local]
        for (int i = tid; i < KC * NCLS; i += 256) {
            int k = i >> 7;          // 0..KC-1
            int n = i & (NCLS - 1);  // 0..127
            s_win [n * LSTR + k] = gate_w [(kc + k) * NCLS + n];
            s_wout[n * LSTR + k] = gate_w [(HIDDEN + kc + k) * NCLS + n];
            s_wt  [n * LSTR + k] = trans_w[(kc + k) * NCLS + n];
        }
        __syncthreads();

#pragma unroll
        for (int k0 = 0; k0 < KC; k0 += 4) {
            // A fragments: lane -> (row = mrow, K = k0+khalf, k0+khalf+1)
            v2f aIn  = *(const v2f*)(ainp  + kc + k0 + khalf);
            v2f aOut = *(const v2f*)(aoutp + kc + k0 + khalf);
#pragma unroll
            for (int t = 0; t < 8; ++t) {
                const int coff = (t * 16 + mrow) * LSTR + k0 + khalf;
                v2f bIn  = *(const v2f*)&s_win [coff];
                v2f bOut = *(const v2f*)&s_wout[coff];
                v2f bT   = *(const v2f*)&s_wt  [coff];
                accG[t] = __builtin_amdgcn_wmma_f32_16x16x4_f32(
                    false, aIn,  false, bIn,  (short)0, accG[t], false, false);
                accG[t] = __builtin_amdgcn_wmma_f32_16x16x4_f32(
                    false, aOut, false, bOut, (short)0, accG[t], false, false);
                accT[t] = __builtin_amdgcn_wmma_f32_16x16x4_f32(
                    false, aOut, false, bT,   (short)0, accT[t], false, false);
            }
        }
    }

    // Epilogue: bias + sigmoid gate, masked store of nodewise readout.
#pragma unroll
    for (int t = 0; t < 8; ++t) {
        const int col = t * 16 + mrow;
        const float gb = gate_b[col];
        const float tb = trans_b[col];
#pragma unroll
        for (int j = 0; j < 8; ++j) {
            int row = waveRow + j + rhalf;       // C layout: VGPR j -> M=j / j+8
            float g    = accG[t][j] + gb;
            float gate = 1.0f / (1.0f + __expf(-g));
            float v    = gate * (accT[t][j] + tb);
            if (row < n_nodes)
                out_node[(size_t)row * NCLS + col] = v;
        }
    }
}

// ---------------------------------------------------------------------------
// Deterministic segment sum over the sorted graph-id list: one block per
// graph; binary search the row range, each thread owns one class column.
// ---------------------------------------------------------------------------
__launch_bounds__(128)
__global__ void segment_sum(const float* __restrict__ node_vals,
                            const int* __restrict__ seg,
                            float* __restrict__ out_graph,
                            int n_nodes)
{
    const int g = blockIdx.x;

    int lo = 0, hi = n_nodes;                    // lower_bound(seg, g)
    while (lo < hi) { int m = (lo + hi) >> 1; if (seg[m] < g) lo = m + 1; else hi = m; }
    int lo2 = lo, hi2 = n_nodes;                 // lower_bound(seg, g+1)
    while (lo2 < hi2) { int m = (lo2 + hi2) >> 1; if (seg[m] < g + 1) lo2 = m + 1; else hi2 = m; }

    const int col = threadIdx.x;
    float s = 0.0f;
    for (int r = lo; r < lo2; ++r)
        s += node_vals[(size_t)r * NCLS + col];
    out_graph[(size_t)g * NCLS + col] = s;
}

// ---------------------------------------------------------------------------
extern "C" void kernel_launch(void* const* d_in, const int* in_sizes, int n_in,
                              void* d_out, int out_size, void* d_ws, size_t ws_size,
                              hipStream_t stream)
{
    const float* Ain     = (const float*)d_in[0];  // raw_node_in  (N,256)
    const float* Aout    = (const float*)d_in[1];  // raw_node_out (N,256)
    const float* gate_w  = (const float*)d_in[2];  // (512,128): Win rows 0..255, Wout 256..511
    const float* gate_b  = (const float*)d_in[3];  // (128,)
    const float* trans_w = (const float*)d_in[4];  // (256,128)
    const float* trans_b = (const float*)d_in[5];  // (128,)
    const int*   seg     = (const int*)d_in[6];    // sorted graph ids (N,)

    const int n_nodes = in_sizes[0] / HIDDEN;
    float* out_node  = (float*)d_out;
    float* out_graph = out_node + (size_t)n_nodes * NCLS;
    const int n_graphs = (out_size - n_nodes * NCLS) / NCLS;

    const int blocks = (n_nodes + BROWS - 1) / BROWS;
    readout_fused<<<blocks, 256, 0, stream>>>(Ain, Aout, gate_w, gate_b,
                                              trans_w, trans_b, out_node, n_nodes);
    segment_sum<<<n_graphs, NCLS, 0, stream>>>(out_node, seg, out_graph, n_nodes);
}